// Memory_61649960567528
// MI455X (gfx1250) — compile-verified
//
#include <hip/hip_runtime.h>
#include <math.h>

typedef __attribute__((ext_vector_type(2))) float v2f;
typedef __attribute__((ext_vector_type(8))) float v8f;

constexpr int B_ = 8, N_ = 2048, W_ = 64, R_ = 4, IN_ = 512;
constexpr float EPSF = 1e-6f;
constexpr int STRIPS = 32;           // 64 link rows per strip

// ---- d_out flat offsets (floats), reference return order ----
constexpr size_t O_RV   = 0;                                   // (B,R,W)
constexpr size_t O_MEM  = O_RV   + (size_t)B_*R_*W_;           // (B,N,W)
constexpr size_t O_LINK = O_MEM  + (size_t)B_*N_*W_;           // (B,1,N,N)
constexpr size_t O_PREC = O_LINK + (size_t)B_*N_*N_;           // (B,1,N)
constexpr size_t O_RW   = O_PREC + (size_t)B_*N_;              // (B,R,N)
constexpr size_t O_WW   = O_RW   + (size_t)B_*R_*N_;           // (B,1,N)
constexpr size_t O_USG  = O_WW   + (size_t)B_*N_;              // (B,N)

// ---- d_ws offsets (floats) ----
constexpr size_t WS_RK   = 0;                 // B*R*W tanh(read keys)
constexpr size_t WS_RS   = WS_RK + 2048;      // B*R softplus
constexpr size_t WS_WK   = WS_RS + 32;        // B*W tanh
constexpr size_t WS_WSTR = WS_WK + 512;       // B softplus
constexpr size_t WS_EV   = WS_WSTR + 8;       // B*W sigmoid
constexpr size_t WS_WV   = WS_EV + 512;       // B*W tanh
constexpr size_t WS_FG   = WS_WV + 512;       // B*R sigmoid
constexpr size_t WS_AG   = WS_FG + 32;        // B sigmoid
constexpr size_t WS_WG   = WS_AG + 8;         // B sigmoid
constexpr size_t WS_RM   = WS_WG + 8;         // B*R*3 softmax
constexpr size_t WS_ALLOC= WS_RM + 96;        // B*N
constexpr size_t WS_CW   = WS_ALLOC + (size_t)B_*N_;        // B*R*N
constexpr size_t WS_FW   = WS_CW   + (size_t)B_*R_*N_;      // B*R*N
constexpr size_t WS_BWP  = WS_FW   + (size_t)B_*R_*N_;      // B*STRIPS*R*N partials

struct GatePtrs { const float* W[10]; const float* bia[10]; };

// Workgroup barrier that only drains DScnt (LDS) — lets global stores keep
// draining asynchronously instead of the compiler's drain-everything barrier.
static __device__ __forceinline__ void lds_barrier() {
  asm volatile("s_wait_dscnt 0x0\n\t"
               "s_barrier_signal -1\n\t"
               "s_barrier_wait -1" ::: "memory");
}

// ---------------- K1: gate projections + activations ----------------
__global__ __launch_bounds__(512) void k_gates(const float* __restrict__ xi,
                                               GatePtrs P, float* __restrict__ ws) {
  __shared__ float xis[IN_];
  __shared__ float rmraw[12];
  const int b = blockIdx.x, tid = threadIdx.x;
  for (int i = tid; i < IN_; i += blockDim.x) xis[i] = xi[b*IN_ + i];
  __syncthreads();
  constexpr int fout[10] = {256,4,64,1,64,64,4,1,1,12};
  constexpr int cum[11]  = {0,256,260,324,325,389,453,457,458,459,471};
  constexpr int act[10]  = {0,1,0,1,2,0,2,2,2,3};   // 0 tanh, 1 softplus, 2 sigmoid, 3 raw(rm)
  constexpr size_t wso[10] = {WS_RK,WS_RS,WS_WK,WS_WSTR,WS_EV,WS_WV,WS_FG,WS_AG,WS_WG,WS_RM};
  const int f = tid;
  if (f < 471) {
    int h = 0;
    while (f >= cum[h+1]) ++h;
    const int row = f - cum[h];
    const float* Wp = P.W[h] + row*IN_;
    float acc = P.bia[h][row];
    for (int k = 0; k < IN_; ++k) acc = fmaf(xis[k], Wp[k], acc);
    float y;
    switch (act[h]) {
      case 0: y = tanhf(acc); break;
      case 1: y = (acc > 20.f) ? acc : log1pf(expf(acc)); break;
      case 2: y = 1.f/(1.f + expf(-acc)); break;
      default: y = acc; break;
    }
    if (h == 9) rmraw[row] = y;
    else ws[wso[h] + (size_t)b*fout[h] + row] = y;
  }
  __syncthreads();
  if (tid < R_) {   // read-mode softmax over groups of 3
    float a0 = rmraw[3*tid], a1 = rmraw[3*tid+1], a2 = rmraw[3*tid+2];
    float m = fmaxf(a0, fmaxf(a1, a2));
    float e0 = expf(a0-m), e1 = expf(a1-m), e2 = expf(a2-m);
    float inv = 1.f/(e0+e1+e2);
    ws[WS_RM + (size_t)b*12 + 3*tid+0] = e0*inv;
    ws[WS_RM + (size_t)b*12 + 3*tid+1] = e1*inv;
    ws[WS_RM + (size_t)b*12 + 3*tid+2] = e2*inv;
  }
}

// ---------------- K2: usage, psi, bitonic argsort, allocation --------
__global__ __launch_bounds__(1024) void k_usage_alloc(
    const float* __restrict__ rwold, const float* __restrict__ wwold,
    const float* __restrict__ uv, float* __restrict__ out, float* __restrict__ ws) {
  __shared__ float key[N_];
  __shared__ int   idx[N_];
  __shared__ float seg[256];
  const int b = blockIdx.x, tid = threadIdx.x;
  float fg[R_];
  for (int r = 0; r < R_; ++r) fg[r] = ws[WS_FG + (size_t)b*R_ + r];
  for (int n = tid; n < N_; n += 1024) {
    float u0 = uv[b*N_ + n];
    float u = u0 + (1.f - u0)*wwold[b*N_ + n];
    float psi = 1.f;
    for (int r = 0; r < R_; ++r) psi *= (1.f - fg[r]*rwold[(size_t)(b*R_+r)*N_ + n]);
    u *= psi;
    out[O_USG + (size_t)b*N_ + n] = u;
    key[n] = EPSF + (1.f - EPSF)*u;
    idx[n] = n;
  }
  __syncthreads();
  // bitonic ascending sort of (key, idx), stable via index tiebreak
  for (int k = 2; k <= N_; k <<= 1) {
    for (int j = k >> 1; j > 0; j >>= 1) {
      for (int t = tid; t < N_; t += 1024) {
        int x = t ^ j;
        if (x > t) {
          float k1 = key[t], k2 = key[x];
          int   i1 = idx[t], i2 = idx[x];
          bool less2 = (k2 < k1) || (k2 == k1 && i2 < i1);
          bool up = ((t & k) == 0);
          if (less2 == up) { key[t]=k2; key[x]=k1; idx[t]=i2; idx[x]=i1; }
        }
      }
      __syncthreads();
    }
  }
  // exclusive cumulative product of sorted keys -> allocation, scatter
  float loc[8]; float p = 1.f;
  if (tid < 256) {
    int base = tid*8;
    for (int k = 0; k < 8; ++k) { loc[k] = p; p *= key[base+k]; }
    seg[tid] = p;
  }
  __syncthreads();
  if (tid == 0) {
    float run = 1.f;
    for (int i = 0; i < 256; ++i) { float t = seg[i]; seg[i] = run; run *= t; }
  }
  __syncthreads();
  if (tid < 256) {
    int base = tid*8;
    float pref = seg[tid];
    for (int k = 0; k < 8; ++k) {
      int i = base + k;
      ws[WS_ALLOC + (size_t)b*N_ + idx[i]] = (1.f - key[i]) * (pref * loc[k]);
    }
  }
}

// ---------------- K3: write content weighting, ww, precedence --------
__global__ __launch_bounds__(256) void k_writew(
    const float* __restrict__ memory, const float* __restrict__ prec_in,
    float* __restrict__ out, float* __restrict__ ws) {
  __shared__ float wk[W_];
  __shared__ float sc[N_];
  __shared__ float red[256];
  const int b = blockIdx.x, tid = threadIdx.x;
  if (tid < W_) wk[tid] = ws[WS_WK + (size_t)b*W_ + tid];
  __syncthreads();
  float kn = 0.f;
  for (int i = 0; i < W_; ++i) kn = fmaf(wk[i], wk[i], kn);
  kn = sqrtf(kn) + EPSF;
  const float beta = ws[WS_WSTR + b];
  const float* mb = memory + (size_t)b*N_*W_;
  for (int it = 0; it < N_/256; ++it) {
    int n = tid + it*256;
    const float* row = mb + (size_t)n*W_;
    float dot = 0.f, sq = 0.f;
    for (int i = 0; i < W_; ++i) { float m = row[i]; dot = fmaf(m, wk[i], dot); sq = fmaf(m, m, sq); }
    sc[n] = dot / ((sqrtf(sq) + EPSF) * kn) * beta;
  }
  __syncthreads();
  float mx = -1e30f;
  for (int it = 0; it < N_/256; ++it) mx = fmaxf(mx, sc[tid + it*256]);
  red[tid] = mx; __syncthreads();
  for (int s = 128; s > 0; s >>= 1) { if (tid < s) red[tid] = fmaxf(red[tid], red[tid+s]); __syncthreads(); }
  mx = red[0]; __syncthreads();
  float sm = 0.f;
  for (int it = 0; it < N_/256; ++it) { int n = tid + it*256; float e = expf(sc[n]-mx); sc[n] = e; sm += e; }
  red[tid] = sm; __syncthreads();
  for (int s = 128; s > 0; s >>= 1) { if (tid < s) red[tid] += red[tid+s]; __syncthreads(); }
  const float inv = 1.f / red[0]; __syncthreads();
  const float ag = ws[WS_AG + b], wg = ws[WS_WG + b];
  float wsum = 0.f;
  for (int it = 0; it < N_/256; ++it) {
    int n = tid + it*256;
    float wcw = sc[n]*inv;
    float w = wg * (ag * ws[WS_ALLOC + (size_t)b*N_ + n] + (1.f - ag)*wcw);
    out[O_WW + (size_t)b*N_ + n] = w;
    sc[n] = w; wsum += w;
  }
  red[tid] = wsum; __syncthreads();
  for (int s = 128; s > 0; s >>= 1) { if (tid < s) red[tid] += red[tid+s]; __syncthreads(); }
  const float tot = red[0]; __syncthreads();
  for (int it = 0; it < N_/256; ++it) {
    int n = tid + it*256;
    out[O_PREC + (size_t)b*N_ + n] = (1.f - tot)*prec_in[b*N_ + n] + sc[n];
  }
}

// ---------------- K4: memory erase/write update ----------------------
__global__ void k_mem(const float* __restrict__ memory,
                      const float* __restrict__ ws, float* __restrict__ out) {
  const int stride = blockDim.x * gridDim.x;
  for (int idx = blockIdx.x*blockDim.x + threadIdx.x; idx < B_*N_*W_; idx += stride) {
    int w = idx & (W_-1);
    int n = (idx >> 6) & (N_-1);
    int b = idx >> 17;
    float wwv = out[O_WW + (size_t)b*N_ + n];
    float ev  = ws[WS_EV + (size_t)b*W_ + w];
    float wv  = ws[WS_WV + (size_t)b*W_ + w];
    out[O_MEM + idx] = memory[idx]*(1.f - wwv*ev) + wwv*wv;
  }
}

// ---------------- K5: read content weighting --------------------------
__global__ __launch_bounds__(256) void k_readw(float* __restrict__ out, float* __restrict__ ws) {
  __shared__ float kk[W_];
  __shared__ float sc[N_];
  __shared__ float red[256];
  const int br = blockIdx.x, b = br / R_, tid = threadIdx.x;
  if (tid < W_) kk[tid] = ws[WS_RK + (size_t)br*W_ + tid];
  __syncthreads();
  float kn = 0.f;
  for (int i = 0; i < W_; ++i) kn = fmaf(kk[i], kk[i], kn);
  kn = sqrtf(kn) + EPSF;
  const float beta = ws[WS_RS + br];
  const float* mb = out + O_MEM + (size_t)b*N_*W_;
  for (int it = 0; it < N_/256; ++it) {
    int n = tid + it*256;
    const float* row = mb + (size_t)n*W_;
    float dot = 0.f, sq = 0.f;
    for (int i = 0; i < W_; ++i) { float m = row[i]; dot = fmaf(m, kk[i], dot); sq = fmaf(m, m, sq); }
    sc[n] = dot / ((sqrtf(sq) + EPSF) * kn) * beta;
  }
  __syncthreads();
  float mx = -1e30f;
  for (int it = 0; it < N_/256; ++it) mx = fmaxf(mx, sc[tid + it*256]);
  red[tid] = mx; __syncthreads();
  for (int s = 128; s > 0; s >>= 1) { if (tid < s) red[tid] = fmaxf(red[tid], red[tid+s]); __syncthreads(); }
  mx = red[0]; __syncthreads();
  float sm = 0.f;
  for (int it = 0; it < N_/256; ++it) { int n = tid + it*256; float e = expf(sc[n]-mx); sc[n] = e; sm += e; }
  red[tid] = sm; __syncthreads();
  for (int s = 128; s > 0; s >>= 1) { if (tid < s) red[tid] += red[tid+s]; __syncthreads(); }
  const float inv = 1.f / red[0];
  for (int it = 0; it < N_/256; ++it) {
    int n = tid + it*256;
    ws[WS_CW + (size_t)br*N_ + n] = sc[n]*inv;
  }
}

// -------- K6: fused link update + fw/bw via V_WMMA_F32_16X16X4_F32 ----
// Workgroup: 4 waves (128 thr); wave rt owns 16-row tile of a 64-row strip.
// Per 16x16 link tile: compute new link (C/D lane layout, branchless diagonal),
// store to d_out, stage in LDS, then 4 K=4 fp32 WMMAs for fw (B = tile^T) and
// 4 for bw (B = tile). A rows 4..15 are junk-by-design (only D rows 0..3 are
// read); bw A operands are jt-invariant and hoisted. Barriers drain DScnt only
// so the 134 MB of link stores flow out asynchronously.
__global__ __launch_bounds__(128) void k_link(
    const float* __restrict__ link_in, const float* __restrict__ prec_in,
    const float* __restrict__ rwold, float* __restrict__ out, float* __restrict__ ws) {
  __shared__ float tile[4][256];
  __shared__ float bwred[4][64];
  const int b = blockIdx.x / STRIPS, s = blockIdx.x % STRIPS;
  const int tid = threadIdx.x;
  const int wave = tid >> 5, lane = tid & 31;
  const int col = lane & 15, hi = lane >> 4;
  const int i0 = s*64, it0 = i0 + wave*16;
  const float* L  = link_in + (size_t)b*N_*N_;
  float*       Lo = out + O_LINK + (size_t)b*N_*N_;
  const float* ww = out + O_WW + (size_t)b*N_;
  const float* pr = prec_in + (size_t)b*N_;
  const float* rw = rwold + (size_t)b*R_*N_;
  const float* rwrow = rw + (size_t)(col & 3)*N_;   // A row (col&3); rows 4..15 junk-ok
  float* tl = tile[wave];
  float wwi[8];
  #pragma unroll
  for (int v = 0; v < 8; ++v) wwi[v] = ww[it0 + v + 8*hi];
  // bw A operands: rw[r, it0 + 4c + 2hi + {0,1}], invariant across jt
  v2f abw[4];
  #pragma unroll
  for (int c = 0; c < 4; ++c) {
    abw[c].x = rwrow[it0 + 4*c + 2*hi + 0];
    abw[c].y = rwrow[it0 + 4*c + 2*hi + 1];
  }
  v8f fwacc = {};
  for (int jt = 0; jt < N_/16; ++jt) {
    const int j0 = jt*16;
    const float wwj = ww[j0 + col];
    const float prj = pr[j0 + col];
    // fw A operands for this j-tile (unpredicated b64 loads, L2/WGP$-hot)
    v2f afw[4];
    #pragma unroll
    for (int c = 0; c < 4; ++c) {
      afw[c].x = rwrow[j0 + 4*c + 2*hi + 0];
      afw[c].y = rwrow[j0 + 4*c + 2*hi + 1];
    }
    #pragma unroll
    for (int v = 0; v < 8; ++v) {          // C/D layout: row = v+8*hi, col = lane&15
      const int gi = it0 + v + 8*hi, gj = j0 + col;
      float lv = L[(size_t)gi*N_ + gj];    // unconditional load
      float nl = (1.f - wwi[v] - wwj)*lv + wwi[v]*prj;
      nl = (gi == gj) ? 0.f : nl;          // branchless diagonal (v_cndmask)
      Lo[(size_t)gi*N_ + gj] = nl;
      tl[(v + 8*hi)*16 + col] = nl;
    }
    lds_barrier();
    v8f bwacc = {};
    #pragma unroll
    for (int c = 0; c < 4; ++c) {
      const int k0 = 4*c + 2*hi;           // A/B layout: K = v + 2*hi within chunk
      // fw[r,i] += rw[r, j0+k] * Lnew[i, j0+k]   (B = tile transposed)
      v2f bbt;
      bbt.x = tl[col*16 + k0 + 0];
      bbt.y = tl[col*16 + k0 + 1];
      fwacc = __builtin_amdgcn_wmma_f32_16x16x4_f32(false, afw[c], false, bbt, (short)0, fwacc, false, false);
      // bw[r,j] += rw[r, it0+k] * Lnew[it0+k, j]  (B = tile)
      v2f bb;
      bb.x = tl[(k0 + 0)*16 + col];
      bb.y = tl[(k0 + 1)*16 + col];
      bwacc = __builtin_amdgcn_wmma_f32_16x16x4_f32(false, abw[c], false, bb, (short)0, bwacc, false, false);
    }
    if (hi == 0) {                          // valid D rows r<4: vgpr v=0..3, lanes 0..15
      #pragma unroll
      for (int v = 0; v < R_; ++v) bwred[wave][v*16 + col] = bwacc[v];
    }
    lds_barrier();
    if (wave == 0 && hi == 0) {             // cross-wave sum -> strip partial
      #pragma unroll
      for (int v = 0; v < R_; ++v) {
        float sum = bwred[0][v*16+col] + bwred[1][v*16+col] + bwred[2][v*16+col] + bwred[3][v*16+col];
        ws[WS_BWP + ((size_t)(b*STRIPS + s)*R_ + v)*N_ + j0 + col] = sum;
      }
    }
    lds_barrier();
  }
  if (hi == 0) {                            // fw complete for this wave's 16 rows
    #pragma unroll
    for (int v = 0; v < R_; ++v)
      ws[WS_FW + (size_t)(b*R_ + v)*N_ + it0 + col] = fwacc[v];
  }
}

// ---------------- K7a: bw partial reduce + read-mode mix --------------
__global__ void k_bw_rw(float* __restrict__ out, const float* __restrict__ ws) {
  const int stride = blockDim.x * gridDim.x;
  for (int i = blockIdx.x*blockDim.x + threadIdx.x; i < B_*R_*N_; i += stride) {
    int n = i & (N_-1);
    int r = (i >> 11) & (R_-1);
    int b = i >> 13;
    float bw = 0.f;
    for (int s2 = 0; s2 < STRIPS; ++s2)
      bw += ws[WS_BWP + ((size_t)(b*STRIPS + s2)*R_ + r)*N_ + n];
    float fw = ws[WS_FW + (size_t)i];
    float cw = ws[WS_CW + (size_t)i];
    const size_t rm = WS_RM + (size_t)(b*R_ + r)*3;
    out[O_RW + (size_t)i] = ws[rm+0]*bw + ws[rm+1]*fw + ws[rm+2]*cw;
  }
}

// ---------------- K7b: read vectors -----------------------------------
__global__ __launch_bounds__(256) void k_readvec(float* __restrict__ out) {
  __shared__ float red[256];
  const int br = blockIdx.x, b = br / R_;
  const int tid = threadIdx.x, w = tid & 63, c = tid >> 6;
  const float* mb  = out + O_MEM + (size_t)b*N_*W_;
  const float* rwp = out + O_RW + (size_t)br*N_;
  float acc = 0.f;
  for (int n = c*(N_/4); n < (c+1)*(N_/4); ++n)
    acc = fmaf(rwp[n], mb[(size_t)n*W_ + w], acc);
  red[tid] = acc;
  __syncthreads();
  if (c == 0)
    out[O_RV + (size_t)br*W_ + w] = red[w] + red[64+w] + red[128+w] + red[192+w];
}

extern "C" void kernel_launch(void* const* d_in, const int* in_sizes, int n_in,
                              void* d_out, int out_size, void* d_ws, size_t ws_size,
                              hipStream_t stream) {
  const float* xi     = (const float*)d_in[0];
  const float* memory = (const float*)d_in[1];
  const float* link   = (const float*)d_in[2];
  const float* prec   = (const float*)d_in[3];
  const float* rwold  = (const float*)d_in[4];
  const float* wwold  = (const float*)d_in[5];
  const float* uv     = (const float*)d_in[6];
  float* out = (float*)d_out;
  float* ws  = (float*)d_ws;
  GatePtrs P;
  for (int h = 0; h < 10; ++h) {
    P.W[h]   = (const float*)d_in[7 + 2*h];
    P.bia[h] = (const float*)d_in[8 + 2*h];
  }
  k_gates      <<<B_,        512, 0, stream>>>(xi, P, ws);
  k_usage_alloc<<<B_,       1024, 0, stream>>>(rwold, wwold, uv, out, ws);
  k_writew     <<<B_,        256, 0, stream>>>(memory, prec, out, ws);
  k_mem        <<<1024,      256, 0, stream>>>(memory, ws, out);
  k_readw      <<<B_*R_,     256, 0, stream>>>(out, ws);
  k_link       <<<B_*STRIPS, 128, 0, stream>>>(link, prec, rwold, out, ws);
  k_bw_rw      <<<256,       256, 0, stream>>>(out, ws);
  k_readvec    <<<B_*R_,     256, 0, stream>>>(out);
}